// FieldPooling_11355893530903
// MI455X (gfx1250) — compile-verified
//
#include <hip/hip_runtime.h>
#include <stdint.h>

// Problem constants (match reference)
#define BATCH   32
#define NPTS    16384
#define CHANS   64
#define NSAMP   4096
#define THREADS 1024
#define PPT     (NPTS / THREADS)   // 16 points per thread, held in VGPRs

// ---------------------------------------------------------------------------
// FPS kernel: one workgroup per batch element (32 waves, wave32).
//  - Stage (x,y,z,f0) of every point into LDS via CDNA5 async-to-LDS loads
//    (ASYNCcnt-tracked, s_wait_asynccnt): 16 B/point => 256 KB of 320 KB LDS.
//  - Points + running min-distance live in VGPRs (64 f32/thread).
//  - Per iteration (4095 total, strictly sequential):
//      * uniform-address LDS broadcast of the centroid float4
//      * exact f32 distance (no FMA contraction -> bit-matches JAX f32),
//        running min, thread-local argmax (f32 compare, first-occurrence)
//      * one u64 key {dist_bits, ~index}: dist>=0 so IEEE bits are order-
//        preserving; ~index makes ties resolve to the SMALLEST index,
//        matching jnp.argmax semantics exactly
//      * 5-step wave32 shfl_xor reduce; lane0 -> parity-double-buffered LDS
//        slot; ONE __syncthreads(); then every wave redundantly reduces the
//        32 slots (conflict-free ds reads) so all threads learn the winner
//        without a second barrier or broadcast cell.
// ---------------------------------------------------------------------------
__global__ __launch_bounds__(THREADS) void fps_kernel(
    const float* __restrict__ in,   // (B, N, 64)
    int* __restrict__ g_idx)        // (B, NSAMP) workspace
{
    extern __shared__ unsigned char smem[];
    float4* s4 = (float4*)smem;                                        // 256 KB
    unsigned long long* s_best = (unsigned long long*)(smem + NPTS * 16); // 2x32 slots

    const int tid  = threadIdx.x;
    const int lane = tid & 31;
    const int wid  = tid >> 5;
    const int bb   = blockIdx.x;

    // ---- Async stage: 16 B per point (x,y,z,feat0), per-lane addresses ----
    const float* src = in + (size_t)bb * NPTS * CHANS;
    uint32_t lds_base = (uint32_t)(uintptr_t)s4;   // low 32 bits of generic = LDS offset
#pragma unroll
    for (int k = 0; k < PPT; ++k) {
        int p = tid + (k << 10);
        uint64_t ga = (uint64_t)(uintptr_t)(src + (size_t)p * CHANS);
        uint32_t la = lds_base + (uint32_t)p * 16u;
        asm volatile("global_load_async_to_lds_b128 %0, %1, off"
                     :: "v"(la), "v"(ga) : "memory");
    }
    asm volatile("s_wait_asynccnt 0x0" ::: "memory");
    __syncthreads();

    // ---- Pull this thread's 16 points into VGPRs ----
    float px[PPT], py[PPT], pz[PPT], pd[PPT];
#pragma unroll
    for (int k = 0; k < PPT; ++k) {
        float4 v = s4[tid + (k << 10)];
        px[k] = v.x; py[k] = v.y; pz[k] = v.z;
        pd[k] = 1.0e10f;                 // reference init value (f32 of 1e10)
    }

    if (tid == 0) g_idx[(size_t)bb * NSAMP] = 0;   // reference emits start index 0

    int cur = 0;
    for (int t = 1; t < NSAMP; ++t) {
        float4 c = s4[cur];              // uniform address -> LDS broadcast
        const float cx = c.x, cy = c.y, cz = c.z;

        float bestd = -1.0f;             // dist >= 0, so k=0 always replaces
        int   bestk = 0;
#pragma unroll
        for (int k = 0; k < PPT; ++k) {
            // Exact f32 (no contraction) to bit-match the JAX reference:
            float dx = __fsub_rn(px[k], cx);
            float dy = __fsub_rn(py[k], cy);
            float dz = __fsub_rn(pz[k], cz);
            float d  = __fadd_rn(__fadd_rn(__fmul_rn(dx, dx), __fmul_rn(dy, dy)),
                                 __fmul_rn(dz, dz));
            float nd = fminf(pd[k], d);
            pd[k] = nd;
            // strict '>' keeps the earliest (smallest) thread-local index
            if (nd > bestd) { bestd = nd; bestk = tid + (k << 10); }
        }

        // One tie-break-exact key per thread; reduce within the wave.
        unsigned long long key =
            ((unsigned long long)__float_as_uint(bestd) << 32) |
            (unsigned int)(~(unsigned int)bestk);
#pragma unroll
        for (int m = 16; m >= 1; m >>= 1) {
            unsigned long long o = __shfl_xor((unsigned long long)key, m, 32);
            key = (o > key) ? o : key;
        }

        // Parity double-buffered slots -> single barrier per iteration.
        const int buf = (t & 1) << 5;
        if (lane == 0) s_best[buf + wid] = key;
        __syncthreads();

        // Every wave redundantly reduces all 32 wave-bests (no 2nd barrier).
        unsigned long long k2 = s_best[buf + lane];
#pragma unroll
        for (int m = 16; m >= 1; m >>= 1) {
            unsigned long long o = __shfl_xor((unsigned long long)k2, m, 32);
            k2 = (o > k2) ? o : k2;
        }
        cur = (int)(~(unsigned int)(k2 & 0xffffffffull));
        if (tid == 0) g_idx[(size_t)bb * NSAMP + t] = cur;
    }
}

// ---------------------------------------------------------------------------
// Gather kernel: out[b, j, :] = in[b, idx[b, j], :].
// concat(points[0:3], feats[3:64]) == the original contiguous 64-float row,
// so this is a pure float4 row copy. Bandwidth bound: ~64 MB total, ~3 us
// at 23.3 TB/s (input rows live in the 192 MB L2 after the FPS pass anyway).
// ---------------------------------------------------------------------------
__global__ void gather_kernel(const float* __restrict__ in,
                              const int* __restrict__ g_idx,
                              float* __restrict__ out)
{
    const int Q = CHANS / 4;                       // 16 float4 per row
    size_t g = (size_t)blockIdx.x * blockDim.x + threadIdx.x;
    size_t total = (size_t)BATCH * NSAMP * Q;
    if (g >= total) return;
    size_t row = g / Q;
    int    q   = (int)(g % Q);
    int    b   = (int)(row >> 12);                 // row / NSAMP
    int    id  = g_idx[row];
    const float4* in4  = (const float4*)in;
    float4*       out4 = (float4*)out;
    out4[row * Q + q] = in4[((size_t)b * NPTS + id) * Q + q];
}

extern "C" void kernel_launch(void* const* d_in, const int* in_sizes, int n_in,
                              void* d_out, int out_size, void* d_ws, size_t ws_size,
                              hipStream_t stream) {
    (void)in_sizes; (void)n_in; (void)out_size; (void)ws_size;
    const float* in  = (const float*)d_in[0];
    int*   idxws     = (int*)d_ws;                 // B*NSAMP ints = 512 KB
    float* out       = (float*)d_out;

    size_t shmem = (size_t)NPTS * 16 + 64 * 8;     // 256 KB pts + 2x32 reduce slots
    fps_kernel<<<BATCH, THREADS, shmem, stream>>>(in, idxws);

    size_t total = (size_t)BATCH * NSAMP * (CHANS / 4);
    int blocks = (int)((total + 255) / 256);
    gather_kernel<<<blocks, 256, 0, stream>>>(in, idxws, out);
}